// Convolution_82514911691296
// MI455X (gfx1250) — compile-verified
//
#include <hip/hip_runtime.h>

typedef float v2f __attribute__((ext_vector_type(2)));
typedef float v8f __attribute__((ext_vector_type(8)));

#define NTAPS 343
#define CENTER_TAP 171           // (3*7+3)*7+3
#define SP 64000                 // 40*40*40
#define WA_TAP 4096              // floats per tap: 4 ocq * 16 icq * 32 lanes * 2
#define IC_STRIDE 1400           // 14*10*10
#define CHUNK 5600               // 4 * 1400  ([4 ic][14 d][10 h][10 w])

#if __has_builtin(__builtin_amdgcn_global_load_async_to_lds_b32) && \
    __has_builtin(__builtin_amdgcn_s_wait_asynccnt)
#define USE_ASYNC 1
#else
#define USE_ASYNC 0
#endif

typedef __attribute__((address_space(1))) int* gint_t;
typedef __attribute__((address_space(3))) int* lint_t;

// ---------------------------------------------------------------------------
// Kernel 1: synthesize conv weights (radial basis x SH tensor product),
// fold the self-connection into the center tap, and write them pre-swizzled
// into the WMMA f32 16x16x4 A-operand lane layout:
//   W_A[tap][ocq][icq][lane][j] = K[oc = ocq*16 + lane%16]
//                                  [ic = icq*4 + (lane>=16 ? 2:0) + j]
// ---------------------------------------------------------------------------
__global__ __launch_bounds__(256) void build_wa_kernel(
    const float* __restrict__ sc_w0,     // [16][16] (u,o)
    const float* __restrict__ sc_w1,     // [16][16] (u,o)
    const float* __restrict__ tp_weight, // [8][1280]
    float* __restrict__ W_A)             // [343][4096]
{
    __shared__ float wsh[5 * 256];   // w_p[u][o]
    __shared__ float Ksh[64 * 64];   // K[oc][ic]

    const int tap = blockIdx.x;
    const int kx = tap / 49;
    const int ky = (tap / 7) % 7;
    const int kz = tap % 7;
    const float fx = (float)(kx - 3);
    const float fy = (float)(ky - 3);
    const float fz = (float)(kz - 3);
    const float r = sqrtf(fx * fx + fy * fy + fz * fz);

    const float step = 3.5f / 9.0f;
    float emb[8];
#pragma unroll
    for (int m = 0; m < 8; ++m) {
        float v = step * (float)(m + 1);
        float diff = (r - v) / step;
        float t1 = diff + 1.0f;
        float t2 = 1.0f - diff;
        float u1 = (t1 > 0.0f) ? __expf(-1.0f / t1) : 0.0f;
        float u2 = (t2 > 0.0f) ? __expf(-1.0f / t2) : 0.0f;
        emb[m] = 1.14136f * 7.3890560989306495f * u1 * u2;  // 1.14136 * e^2
    }

    float sh1[3] = {0.0f, 0.0f, 0.0f};
    if (r > 0.0f) {
        float s = 1.7320508075688772f / r;  // sqrt(3)/r
        sh1[0] = s * fx; sh1[1] = s * fy; sh1[2] = s * fz;
    }

    const int tid = threadIdx.x;
    {
        int u = tid >> 4, o = tid & 15;
#pragma unroll
        for (int p = 0; p < 5; ++p) {
            float a = 0.0f;
#pragma unroll
            for (int m = 0; m < 8; ++m)
                a += emb[m] * tp_weight[m * 1280 + (p * 16 + u) * 16 + o];
            wsh[p * 256 + tid] = a * (1.0f / 343.0f);
        }
    }
    __syncthreads();

    const float c1 = 0.17677669529663687f;  // sqrt(1/32)
    const float c2 = 0.10206207261596575f;  // c1/sqrt(3)
    const float c3 = 0.14433756729740643f;  // 0.25/sqrt(3)
    const float c4 = 0.14433756729740643f;
    const float c5 = 0.10206207261596575f;  // 0.25/sqrt(6)
    const float inv = 0.25f;                // 1/sqrt(16)
    const bool center = (tap == CENTER_TAP);

    for (int idx = tid; idx < 4096; idx += 256) {
        int oc = idx >> 6, ic = idx & 63;
        float val;
        if (oc < 16) {
            int o = oc;
            if (ic < 16) {                       // scalar -> scalar
                int u = ic;
                val = c1 * wsh[0 * 256 + u * 16 + o];
                if (center) val += inv * sc_w0[u * 16 + o];
            } else {                             // vector -> scalar
                int u = (ic - 16) / 3, i = (ic - 16) % 3;
                val = c2 * sh1[i] * wsh[1 * 256 + u * 16 + o];
            }
        } else {
            int o = (oc - 16) / 3, k = (oc - 16) % 3;
            if (ic < 16) {                       // scalar -> vector
                int u = ic;
                val = c3 * sh1[k] * wsh[2 * 256 + u * 16 + o];
            } else {                             // vector -> vector
                int u = (ic - 16) / 3, i = (ic - 16) % 3;
                if (i == k) {
                    val = c4 * wsh[3 * 256 + u * 16 + o];
                    if (center) val += inv * sc_w1[u * 16 + o];
                } else {
                    int j = 3 - i - k;
                    float sign = (k == ((i + 2) % 3)) ? 1.0f : -1.0f;
                    val = c5 * sign * sh1[j] * wsh[4 * 256 + u * 16 + o];
                }
            }
        }
        Ksh[idx] = val;
    }
    __syncthreads();

    float* wa = W_A + (size_t)tap * WA_TAP;
    for (int idx = tid; idx < 4096; idx += 256) {
        int j    = idx & 1;
        int lane = (idx >> 1) & 31;
        int icq  = (idx >> 6) & 15;
        int ocq  = idx >> 10;
        int oc = ocq * 16 + (lane & 15);
        int ic = icq * 4 + ((lane >> 4) << 1) + j;
        wa[idx] = Ksh[oc * 64 + ic];
    }
}

// ---------------------------------------------------------------------------
// Kernel 2: implicit-GEMM conv on the WMMA pipe.
// Block = 256 threads = 8 waves. Tile: 64 out-ch x (8 d-planes x 4h x 4w).
// Wave w owns d-plane d0+w and a 4x4 (h,w) patch = 16 N positions, all 4
// oc-tiles (32 acc VGPRs). 40 % 4 == 0 and 40 % 8 == 0, so every WMMA lane
// produces a stored output (no masking, EXEC all-ones throughout).
// Input halo for a 4-ic chunk lives in LDS ([ic][d][h][w], double-buffered);
// next chunk is prefetched with GLOBAL_LOAD_ASYNC_TO_LDS_B32 while the tap
// loop runs (ASYNCcnt + barrier hand-off). OOB halo slots are zeroed once
// and never re-written (OOB-ness is chunk-invariant), so async loads need
// no zero-fill predication.
// ---------------------------------------------------------------------------
__global__ __launch_bounds__(256) void conv_wmma_kernel(
    const float* __restrict__ x,    // [2][64][40][40][40]
    const float* __restrict__ W_A,  // [343][4096]
    float* __restrict__ out)        // [2][64][40][40][40]
{
    __shared__ float lds[2 * CHUNK];

    const int tid  = threadIdx.x;
    const int lane = tid & 31;
    const int wav  = tid >> 5;       // 0..7 -> d plane within tile
    const int n    = lane & 15;      // N index: (hh = n>>2, ww = n&3)
    const int hi16 = lane >> 4;      // 0/1
    const int icl  = hi16 << 1;      // K base within chunk: 0 or 2

    const int w0 = blockIdx.x * 4;           // 0..36
    const int h0 = (blockIdx.y % 10) * 4;    // 0..36
    const int d0 = (blockIdx.y / 10) * 8;    // 0..32
    const int b  = blockIdx.z;

    const float* xb = x + (size_t)b * 64 * SP;

    // per-lane LDS base for B reads; tap adds kx*100 + ky*10 + kz
    const int lbase = icl * IC_STRIDE + wav * 100 + (n >> 2) * 10 + (n & 3);

    v8f acc0 = {}; v8f acc1 = {}; v8f acc2 = {}; v8f acc3 = {};

#if USE_ASYNC
    // zero both buffers once; OOB halo slots stay zero for the whole kernel
    for (int idx = tid; idx < 2 * CHUNK; idx += 256) lds[idx] = 0.0f;
    __syncthreads();

    // stage chunk `icc` (4 input channels) into buffer `buf` via async copies
    auto stage_async = [&](int icc, int buf) {
        const float* src = xb + (size_t)icc * 4 * SP;
#pragma unroll 1
        for (int k = 0; k < 22; ++k) {
            int idx = tid + k * 256;
            if (idx < CHUNK) {
                int ww = idx % 10;
                int t  = idx / 10;
                int hh = t % 10; t /= 10;
                int dd = t % 14;
                int ic = t / 14;
                int d = d0 + dd - 3, h = h0 + hh - 3, w = w0 + ww - 3;
                if ((unsigned)d < 40u && (unsigned)h < 40u && (unsigned)w < 40u) {
                    __builtin_amdgcn_global_load_async_to_lds_b32(
                        (gint_t)(src + (size_t)ic * SP + (d * 40 + h) * 40 + w),
                        (lint_t)(&lds[buf * CHUNK + idx]),
                        0, 0);
                }
            }
        }
    };

    stage_async(0, 0);
    __builtin_amdgcn_s_wait_asynccnt(0);
    __syncthreads();
#endif

    for (int icc = 0; icc < 16; ++icc) {   // 16 chunks of 4 input channels
#if USE_ASYNC
        const int bufbase = (icc & 1) * CHUNK;
        if (icc < 15) stage_async(icc + 1, (icc + 1) & 1);
#else
        const int bufbase = 0;
        __syncthreads();
        for (int idx = tid; idx < CHUNK; idx += 256) {
            int ww = idx % 10;
            int t  = idx / 10;
            int hh = t % 10; t /= 10;
            int dd = t % 14;
            int ic = t / 14;
            int d = d0 + dd - 3, h = h0 + hh - 3, w = w0 + ww - 3;
            float v = 0.0f;
            if ((unsigned)d < 40u && (unsigned)h < 40u && (unsigned)w < 40u)
                v = xb[(size_t)(icc * 4 + ic) * SP + (d * 40 + h) * 40 + w];
            lds[idx] = v;
        }
        __syncthreads();
#endif

        const float* ap_base = W_A + icc * 64 + lane * 2;
        for (int kx = 0; kx < 7; ++kx)
        for (int ky = 0; ky < 7; ++ky) {
            const float* ap = ap_base + (size_t)((kx * 7 + ky) * 7) * WA_TAP;
            const int lrow = bufbase + lbase + kx * 100 + ky * 10;
#pragma unroll
            for (int kz = 0; kz < 7; ++kz) {
                v2f bv;
                bv.x = lds[lrow + kz];
                bv.y = lds[lrow + kz + IC_STRIDE];
                v2f a0 = *(const v2f*)(ap + (size_t)kz * WA_TAP);
                v2f a1 = *(const v2f*)(ap + (size_t)kz * WA_TAP + 1024);
                v2f a2 = *(const v2f*)(ap + (size_t)kz * WA_TAP + 2048);
                v2f a3 = *(const v2f*)(ap + (size_t)kz * WA_TAP + 3072);
                acc0 = __builtin_amdgcn_wmma_f32_16x16x4_f32(
                    false, a0, false, bv, (short)0, acc0, false, false);
                acc1 = __builtin_amdgcn_wmma_f32_16x16x4_f32(
                    false, a1, false, bv, (short)0, acc1, false, false);
                acc2 = __builtin_amdgcn_wmma_f32_16x16x4_f32(
                    false, a2, false, bv, (short)0, acc2, false, false);
                acc3 = __builtin_amdgcn_wmma_f32_16x16x4_f32(
                    false, a3, false, bv, (short)0, acc3, false, false);
            }
        }

#if USE_ASYNC
        __builtin_amdgcn_s_wait_asynccnt(0);
        __syncthreads();
#endif
    }

    // store: C/D layout -> lane l, VGPR r holds (M = r + (l>=16 ? 8:0), N = l%16)
    {
        const int d = d0 + wav;
        const int h = h0 + (n >> 2);
        const int w = w0 + (n & 3);
        const size_t sp = (size_t)(d * 40 + h) * 40 + w;
        float* ob = out + (size_t)b * 64 * SP + sp;
        const int mo = hi16 * 8;
#pragma unroll
        for (int r = 0; r < 8; ++r) {
            ob[(size_t)( 0 + r + mo) * SP] = acc0[r];
            ob[(size_t)(16 + r + mo) * SP] = acc1[r];
            ob[(size_t)(32 + r + mo) * SP] = acc2[r];
            ob[(size_t)(48 + r + mo) * SP] = acc3[r];
        }
    }
}

extern "C" void kernel_launch(void* const* d_in, const int* in_sizes, int n_in,
                              void* d_out, int out_size, void* d_ws, size_t ws_size,
                              hipStream_t stream) {
    // setup_inputs order: x, sc_w0, sc_w1, tp_weight (all float32)
    const float* x    = (const float*)d_in[0];
    const float* scw0 = (const float*)d_in[1];
    const float* scw1 = (const float*)d_in[2];
    const float* tpw  = (const float*)d_in[3];
    float* W_A = (float*)d_ws;  // needs 343*4096*4 = 5,619,712 bytes of d_ws

    build_wa_kernel<<<NTAPS, 256, 0, stream>>>(scw0, scw1, tpw, W_A);
    // grid: x = 10 w-tiles, y = 10 h-tiles * 5 d-tiles, z = batch
    conv_wmma_kernel<<<dim3(10, 50, 2), 256, 0, stream>>>(x, W_A, (float*)d_out);
}